// Link_21698174780141
// MI455X (gfx1250) — compile-verified
//
#include <hip/hip_runtime.h>

typedef __attribute__((ext_vector_type(2))) float v2f;
typedef __attribute__((ext_vector_type(8))) float v8f;

#define N_STATES 64
#define N_EDGES  128
#define BATCH    524288
#define KTOT     68        // 64 in_prob cols + 2 llrs cols + 2 zero-pad cols
#define NPAIRS   (KTOT / 2)        // 34 K-row pairs
#define PSTRIDE  144               // p-row stride in 8-byte entries (288 floats):
                                   // lane-half delta = 1152 B = +32 banks -> conflict-free
#define ROWS_PER_WAVE   16
#define WAVES_PER_BLOCK 8
#define ROWS_PER_BLOCK  (ROWS_PER_WAVE * WAVES_PER_BLOCK)   // 128

__global__ __launch_bounds__(256)
void viterbi_acs_wmma(const float* __restrict__ in_prob,   // [B,64]
                      const float* __restrict__ llrs,      // [B,2]
                      const float* __restrict__ s2e,       // [64,128]
                      const float* __restrict__ mask,      // [64,128]
                      const float* __restrict__ l2e,       // [2,128]
                      float* __restrict__ out)             // [B,64] max ++ [B,64] idx
{
    // K-pair interleaved B operand: Bsw[p][c] = { Bcomb[2p][c], Bcomb[2p+1][c] }
    // so one 8-byte LDS load yields a complete v2f WMMA B operand per lane.
    __shared__ float Bsw[NPAIRS * PSTRIDE * 2];

    const int tid = threadIdx.x;

    // ---- Stage combined weight matrix Bcomb[68][128] (swizzled) into LDS ----
    // rows 0..63 : states_to_edges * mask
    // rows 64..65: llrs_to_edges   (folds branch metrics into the same GEMM)
    // rows 66..67: zero pad        (K multiple of 4)
    for (int idx = tid; idx < N_STATES * N_EDGES; idx += 256) {
        const int r = idx >> 7, c = idx & 127;
        Bsw[(r >> 1) * (PSTRIDE * 2) + c * 2 + (r & 1)] = s2e[idx] * mask[idx];
    }
    if (tid < 2 * N_EDGES) {
        const int r = tid >> 7, c = tid & 127;
        Bsw[32 * (PSTRIDE * 2) + c * 2 + r] = l2e[tid];   // K rows 64,65
        Bsw[33 * (PSTRIDE * 2) + c * 2 + r] = 0.0f;       // K rows 66,67 (pad)
    }
    __syncthreads();

    const int wave  = tid >> 5;
    const int lane  = tid & 31;
    const int lhalf = lane & 15;
    const int hsel  = lane >> 4;          // 0: K {0,1}; 1: K {2,3} within a 4-slice
    const int m0    = (blockIdx.x * WAVES_PER_BLOCK + wave) * ROWS_PER_WAVE;
    const int mA    = m0 + lhalf;         // A-matrix row owned by this lane
    const float* __restrict__ aRow = in_prob + (size_t)mA * 64 + 2 * hsel;

    v8f acc[8] = {};                      // 8 column tiles of 16 -> 128 edges

    #pragma unroll
    for (int k = 0; k < KTOT / 4; ++k) {
        // A operand: 16x4 f32 slice (lane = M, lane half selects K pair)
        v2f a;
        if (k < 16) {
            const float2 av = *(const float2*)(aRow + 4 * k);
            a.x = av.x; a.y = av.y;
        } else {
            if (lane < 16) {   // K = 64,65 -> llrs ; K = 66,67 -> zero pad
                const float2 lv = *(const float2*)(llrs + (size_t)mA * 2);
                a.x = lv.x; a.y = lv.y;
            } else {
                a.x = 0.0f; a.y = 0.0f;
            }
        }
        // B operands: pair index p = 2k + hsel, one b64 LDS load per tile
        const float* __restrict__ bBase =
            Bsw + (2 * k + hsel) * (PSTRIDE * 2) + lhalf * 2;
        #pragma unroll
        for (int n = 0; n < 8; ++n) {     // 8 tiles of 16 edge columns
            const v2f b = *(const v2f*)(bBase + 32 * n);
            acc[n] = __builtin_amdgcn_wmma_f32_16x16x4_f32(
                         false, a, false, b, (short)0, acc[n], false, false);
        }
    }

    // ---- ACS epilogue: pairwise compare-select over adjacent edge columns ----
    // D layout: VGPR i holds M=i (lanes 0-15) / M=8+i (lanes 16-31), N = lane&15.
    const bool   odd    = (lane & 1) != 0;
    const int    rowAdd = hsel ? 8 : 0;
    const size_t TOT    = (size_t)BATCH * N_STATES;

    #pragma unroll
    for (int n = 0; n < 8; ++n) {
        const int d = 8 * n + (lhalf >> 1);           // dest state for this lane pair
        #pragma unroll
        for (int i = 0; i < 8; ++i) {
            const float v     = acc[n][i];
            const float other = __shfl_xor(v, 1, 32); // competing edge metric
            // even lane: max value; odd lane: argmax (0/1, ties -> 0 like jnp.argmax)
            const float outv  = odd ? ((v > other) ? 1.0f : 0.0f) : fmaxf(v, other);
            const int   m     = m0 + i + rowAdd;
            out[(odd ? TOT : 0) + (size_t)m * N_STATES + d] = outv;
        }
    }
}

extern "C" void kernel_launch(void* const* d_in, const int* in_sizes, int n_in,
                              void* d_out, int out_size, void* d_ws, size_t ws_size,
                              hipStream_t stream) {
    (void)in_sizes; (void)n_in; (void)d_ws; (void)ws_size; (void)out_size;
    const float* in_prob = (const float*)d_in[0];
    const float* llrs    = (const float*)d_in[1];
    const float* s2e     = (const float*)d_in[2];
    const float* mask    = (const float*)d_in[3];
    const float* l2e     = (const float*)d_in[4];
    float* out           = (float*)d_out;

    dim3 grid(BATCH / ROWS_PER_BLOCK);   // 4096 blocks
    dim3 block(256);                     // 8 wave32 per block
    viterbi_acs_wmma<<<grid, block, 0, stream>>>(in_prob, llrs, s2e, mask, l2e, out);
}